// ControlBasedAggregation_12292196401207
// MI455X (gfx1250) — compile-verified
//
#include <hip/hip_runtime.h>

// ---------------------------------------------------------------------------
// ControlBasedAggregation for gfx1250 (MI455X).
// Pipeline: WMMA-accelerated edge affinities -> mean-threshold mask ->
// masked CSR -> 30x label propagation -> unique/compact -> max-pool ->
// bitonic-sorted coalesced coarse edges.
// ---------------------------------------------------------------------------

typedef __attribute__((ext_vector_type(2))) float v2f;
typedef __attribute__((ext_vector_type(8))) float v8f;

#define D_FEAT 64

__device__ __forceinline__ unsigned fenc(float f) {
  unsigned u = __float_as_uint(f);
  return (u & 0x80000000u) ? ~u : (u | 0x80000000u);
}
__device__ __forceinline__ float fdec(unsigned u) {
  return (u & 0x80000000u) ? __uint_as_float(u & 0x7FFFFFFFu)
                           : __uint_as_float(~u);
}

// ---- per-node squared norms -------------------------------------------------
__global__ void node_sq_kernel(const float* __restrict__ x,
                               float* __restrict__ sq, int n) {
  int v = blockIdx.x * blockDim.x + threadIdx.x;
  if (v >= n) return;
  const float4* p = (const float4*)(x + (size_t)v * D_FEAT);
  float s = 0.f;
#pragma unroll
  for (int i = 0; i < D_FEAT / 4; ++i) {
    float4 q = p[i];
    s += q.x * q.x + q.y * q.y + q.z * q.z + q.w * q.w;
  }
  sq[v] = s;
}

// ---- edge affinities via V_WMMA_F32_16X16X4_F32 ----------------------------
// One wave computes 16 edge dot-products: A = 16 gathered src rows (16x64),
// B = 16 gathered dst rows as columns (64x16), accumulate 16 K=4 steps,
// extract the diagonal of the 16x16 f32 result via LDS.
__global__ void edge_aff_wmma_kernel(const float* __restrict__ x,
                                     const int* __restrict__ row,
                                     const int* __restrict__ col,
                                     const float* __restrict__ sq,
                                     float* __restrict__ aff, int E) {
  __shared__ float tile[8 * 256];  // 8 waves/block, 256 f32 tile each
  int lane = threadIdx.x & 31;
  int wib = threadIdx.x >> 5;
  int wave = blockIdx.x * (blockDim.x >> 5) + wib;
  int m = lane & 15;
  int e = wave * 16 + m;
  bool ev = (e < E);
  int r = ev ? row[e] : 0;
  int c = ev ? col[e] : 0;
  const float* xr = x + (size_t)r * D_FEAT;
  const float* xc = x + (size_t)c * D_FEAT;
  // A 16x4 f32 layout: lanes 0-15 hold K=0,1 in v[0],v[1]; lanes 16-31 K=2,3.
  int kh = (lane >> 4) << 1;  // 0 or 2
  v8f acc = {};
#pragma unroll
  for (int k0 = 0; k0 < D_FEAT; k0 += 4) {
    v2f a, b;
    a.x = xr[k0 + kh];
    a.y = xr[k0 + kh + 1];
    b.x = xc[k0 + kh];
    b.y = xc[k0 + kh + 1];
    acc = __builtin_amdgcn_wmma_f32_16x16x4_f32(false, a, false, b, (short)0,
                                                acc, false, false);
  }
  float* t = tile + wib * 256;
#pragma unroll
  for (int g = 0; g < 8; ++g) t[lane * 8 + g] = acc[g];
  __syncthreads();
  if (lane < 16 && ev) {
    // D[m][n]: m<8 -> lane n, vgpr m ; m>=8 -> lane n+16, vgpr m-8
    float d = (lane < 8) ? t[lane * 8 + lane] : t[(lane + 16) * 8 + (lane - 8)];
    float v = sq[r] + sq[c] - 2.0f * d;
    aff[e] = sqrtf(fmaxf(v, 0.0f));
  }
}

// ---- scatter-mean threshold + mask + masked in-degree ----------------------
__global__ void edge_degsum_kernel(const int* __restrict__ row,
                                   const float* __restrict__ aff,
                                   float* __restrict__ sumaff,
                                   float* __restrict__ degf, int E) {
  int e = blockIdx.x * blockDim.x + threadIdx.x;
  if (e >= E) return;
  int r = row[e];
  atomicAdd(&sumaff[r], aff[e]);
  atomicAdd(&degf[r], 1.0f);
}

__global__ void node_mean_kernel(const float* __restrict__ sumaff,
                                 const float* __restrict__ degf,
                                 float* __restrict__ mean, int n) {
  int v = blockIdx.x * blockDim.x + threadIdx.x;
  if (v >= n) return;
  mean[v] = sumaff[v] / fmaxf(degf[v], 1.0f);
}

__global__ void edge_mask_kernel(const int* __restrict__ row,
                                 const int* __restrict__ col,
                                 const float* __restrict__ aff,
                                 const float* __restrict__ mean,
                                 int* __restrict__ mask,
                                 int* __restrict__ deg_in, int E) {
  int e = blockIdx.x * blockDim.x + threadIdx.x;
  if (e >= E) return;
  int r = row[e], c = col[e];
  int mk = (aff[e] <= fminf(mean[r], mean[c])) ? 1 : 0;
  mask[e] = mk;
  if (mk) atomicAdd(&deg_in[c], 1);
}

// ---- single-block exclusive scan (n up to ~64K) ----------------------------
__global__ void exscan_kernel(const int* __restrict__ in, int* __restrict__ out,
                              int n) {
  __shared__ int part[1024];
  int tid = threadIdx.x, T = blockDim.x;
  int chunk = (n + T - 1) / T;
  int s = tid * chunk;
  int epos = min(s + chunk, n);
  int sum = 0;
  for (int i = s; i < epos; ++i) sum += in[i];
  part[tid] = sum;
  __syncthreads();
  for (int off = 1; off < T; off <<= 1) {
    int v = part[tid];
    int add = (tid >= off) ? part[tid - off] : 0;
    __syncthreads();
    part[tid] = v + add;
    __syncthreads();
  }
  int run = (tid > 0) ? part[tid - 1] : 0;
  for (int i = s; i < epos; ++i) {
    out[i] = run;
    run += in[i];
  }
  if (tid == 0) out[n] = part[T - 1];
}

__global__ void csr_scatter_kernel(const int* __restrict__ row,
                                   const int* __restrict__ col,
                                   const int* __restrict__ mask,
                                   int* __restrict__ cursor,
                                   int* __restrict__ csr_src, int E) {
  int e = blockIdx.x * blockDim.x + threadIdx.x;
  if (e >= E) return;
  if (!mask[e]) return;
  int p = atomicAdd(&cursor[col[e]], 1);
  csr_src[p] = row[e];
}

// ---- label propagation -----------------------------------------------------
__global__ void iota_kernel(int* __restrict__ lab, int n) {
  int v = blockIdx.x * blockDim.x + threadIdx.x;
  if (v < n) lab[v] = v;
}

__global__ void lp_iter_kernel(const int* __restrict__ rp,
                               const int* __restrict__ src,
                               const int* __restrict__ lin,
                               int* __restrict__ lout, int n) {
  int v = blockIdx.x * blockDim.x + threadIdx.x;
  if (v >= n) return;
  int s = rp[v], e = rp[v + 1];
  if (s == e) {
    lout[v] = lin[v];
    return;
  }
  long long best = -1;
  for (int i = s; i < e; ++i) {
    int li = lin[src[i]];
    int cnt = 0;
    for (int j = s; j < e; ++j) cnt += (lin[src[j]] == li) ? 1 : 0;
    long long score = (long long)cnt * n + (long long)(n - 1 - li);
    if (score > best) best = score;
  }
  lout[v] = n - 1 - (int)(best % n);
}

// ---- unique/compact --------------------------------------------------------
__global__ void mark_present_kernel(const int* __restrict__ lab,
                                    int* __restrict__ present, int n) {
  int v = blockIdx.x * blockDim.x + threadIdx.x;
  if (v < n) present[lab[v]] = 1;
}

__global__ void assign_cluster_kernel(const int* __restrict__ lab,
                                      const int* __restrict__ exs,
                                      int* __restrict__ cluster, int n) {
  int v = blockIdx.x * blockDim.x + threadIdx.x;
  if (v < n) cluster[v] = exs[lab[v]];
}

// ---- max pooling -----------------------------------------------------------
__global__ void pool_node_kernel(const int* __restrict__ cluster,
                                 const int* __restrict__ batch,
                                 int* __restrict__ csize,
                                 int* __restrict__ cbmax, int n) {
  int v = blockIdx.x * blockDim.x + threadIdx.x;
  if (v >= n) return;
  int c = cluster[v];
  atomicAdd(&csize[c], 1);
  atomicMax(&cbmax[c], batch[v]);
}

__global__ void pool_feat_kernel(const float* __restrict__ x,
                                 const int* __restrict__ cluster,
                                 unsigned* __restrict__ cmax, int n) {
  int idx = blockIdx.x * blockDim.x + threadIdx.x;
  if (idx >= n * D_FEAT) return;
  int v = idx >> 6;
  int f = idx & 63;
  atomicMax(&cmax[cluster[v] * D_FEAT + f], fenc(x[idx]));
}

__global__ void write_cx_kernel(const unsigned* __restrict__ cmax,
                                const int* __restrict__ csize,
                                float* __restrict__ o_cx, int n) {
  int idx = blockIdx.x * blockDim.x + threadIdx.x;
  if (idx >= n * D_FEAT) return;
  int c = idx >> 6;
  o_cx[idx] = (csize[c] > 0) ? fdec(cmax[idx]) : 0.0f;
}

__global__ void write_cb_cluster_kernel(const int* __restrict__ cbmax,
                                        const int* __restrict__ cluster,
                                        float* __restrict__ o_cb,
                                        float* __restrict__ o_cl, int n) {
  int v = blockIdx.x * blockDim.x + threadIdx.x;
  if (v >= n) return;
  o_cb[v] = (float)cbmax[v];
  o_cl[v] = (float)cluster[v];
}

// ---- coarse-edge coalescing: keys, bitonic sort, emit ----------------------
__global__ void build_keys_kernel(const int* __restrict__ row,
                                  const int* __restrict__ col,
                                  const int* __restrict__ mask,
                                  const int* __restrict__ cluster,
                                  int* __restrict__ keys, int E, int n, int M) {
  int i = blockIdx.x * blockDim.x + threadIdx.x;
  if (i >= M) return;
  if (i < E) {
    keys[i] = mask[i] ? (cluster[row[i]] * n + cluster[col[i]]) : (n * n);
  } else {
    keys[i] = 0x7FFFFFFF;  // pad sorts after sentinel n*n
  }
}

__global__ void bitonic_kernel(int* __restrict__ keys, int k, int j) {
  int i = blockIdx.x * blockDim.x + threadIdx.x;
  int ixj = i ^ j;
  if (ixj > i) {
    int a = keys[i], b = keys[ixj];
    bool up = ((i & k) == 0);
    bool sw = up ? (a > b) : (a < b);
    if (sw) {
      keys[i] = b;
      keys[ixj] = a;
    }
  }
}

__global__ void emit_edges_kernel(const int* __restrict__ keys,
                                  float* __restrict__ orow,
                                  float* __restrict__ ocol, int E, int n) {
  int p = blockIdx.x * blockDim.x + threadIdx.x;
  if (p >= E) return;
  int kk = keys[p];
  int sent = n * n;
  bool valid = (kk < sent) && (p == 0 || keys[p - 1] != kk);
  orow[p] = valid ? (float)(kk / n) : (float)n;
  ocol[p] = valid ? (float)(kk % n) : (float)n;
}

// ---------------------------------------------------------------------------
extern "C" void kernel_launch(void* const* d_in, const int* in_sizes, int n_in,
                              void* d_out, int out_size, void* d_ws,
                              size_t ws_size, hipStream_t stream) {
  (void)n_in;
  (void)out_size;
  (void)ws_size;
  const float* x = (const float*)d_in[0];
  const int* ei = (const int*)d_in[1];
  const int* batch = (const int*)d_in[2];
  int n = in_sizes[2];
  int E = in_sizes[1] / 2;
  const int* row = ei;
  const int* col = ei + E;

  // bump-allocate scratch
  char* wsb = (char*)d_ws;
  size_t off = 0;
  auto alloc = [&](size_t bytes) -> void* {
    void* p = wsb + off;
    off += (bytes + 255) & ~(size_t)255;
    return p;
  };
  float* sq = (float*)alloc((size_t)n * 4);
  float* aff = (float*)alloc((size_t)E * 4);
  float* sumaff = (float*)alloc((size_t)n * 4);
  float* degf = (float*)alloc((size_t)n * 4);
  float* mean = (float*)alloc((size_t)n * 4);
  int* mask = (int*)alloc((size_t)E * 4);
  int* deg_in = (int*)alloc((size_t)(n + 1) * 4);
  int* row_ptr = (int*)alloc((size_t)(n + 1) * 4);
  int* cursor = (int*)alloc((size_t)(n + 1) * 4);
  int* csr_src = (int*)alloc((size_t)E * 4);
  int* labA = (int*)alloc((size_t)n * 4);
  int* labB = (int*)alloc((size_t)n * 4);
  int* present = (int*)alloc((size_t)n * 4);
  int* exs = (int*)alloc((size_t)(n + 1) * 4);
  int* cluster = (int*)alloc((size_t)n * 4);
  int* csize = (int*)alloc((size_t)n * 4);
  int* cbmax = (int*)alloc((size_t)n * 4);
  unsigned* cmax = (unsigned*)alloc((size_t)n * D_FEAT * 4);
  int M = 1;
  while (M < E) M <<= 1;  // 2^21 for E=1.28M
  int* keys = (int*)alloc((size_t)M * 4);

  float* o = (float*)d_out;
  float* o_cx = o;
  float* o_er = o + (size_t)n * D_FEAT;
  float* o_ec = o_er + E;
  float* o_cb = o_ec + E;
  float* o_cl = o_cb + n;

  const int B = 256;
  int gN = (n + B - 1) / B;
  int gE = (E + B - 1) / B;
  int gNF = (n * D_FEAT + B - 1) / B;
  int gM = (M + B - 1) / B;

  hipMemsetAsync(sumaff, 0, (size_t)n * 4, stream);
  hipMemsetAsync(degf, 0, (size_t)n * 4, stream);
  hipMemsetAsync(deg_in, 0, (size_t)(n + 1) * 4, stream);
  hipMemsetAsync(present, 0, (size_t)n * 4, stream);
  hipMemsetAsync(csize, 0, (size_t)n * 4, stream);
  hipMemsetAsync(cbmax, 0, (size_t)n * 4, stream);
  hipMemsetAsync(cmax, 0, (size_t)n * D_FEAT * 4, stream);  // 0 == -inf enc

  node_sq_kernel<<<gN, B, 0, stream>>>(x, sq, n);
  int waves = (E + 15) / 16;
  int gW = (waves + 7) / 8;  // 8 waves per 256-thread block
  edge_aff_wmma_kernel<<<gW, B, 0, stream>>>(x, row, col, sq, aff, E);
  edge_degsum_kernel<<<gE, B, 0, stream>>>(row, aff, sumaff, degf, E);
  node_mean_kernel<<<gN, B, 0, stream>>>(sumaff, degf, mean, n);
  edge_mask_kernel<<<gE, B, 0, stream>>>(row, col, aff, mean, mask, deg_in, E);
  exscan_kernel<<<1, 1024, 0, stream>>>(deg_in, row_ptr, n);
  hipMemcpyAsync(cursor, row_ptr, (size_t)n * 4, hipMemcpyDeviceToDevice,
                 stream);
  csr_scatter_kernel<<<gE, B, 0, stream>>>(row, col, mask, cursor, csr_src, E);

  iota_kernel<<<gN, B, 0, stream>>>(labA, n);
  int* lin = labA;
  int* lout = labB;
  for (int it = 0; it < 30; ++it) {
    lp_iter_kernel<<<gN, B, 0, stream>>>(row_ptr, csr_src, lin, lout, n);
    int* t = lin;
    lin = lout;
    lout = t;
  }
  // 30 swaps -> final labels back in labA (== lin)

  mark_present_kernel<<<gN, B, 0, stream>>>(lin, present, n);
  exscan_kernel<<<1, 1024, 0, stream>>>(present, exs, n);
  assign_cluster_kernel<<<gN, B, 0, stream>>>(lin, exs, cluster, n);

  pool_node_kernel<<<gN, B, 0, stream>>>(cluster, batch, csize, cbmax, n);
  pool_feat_kernel<<<gNF, B, 0, stream>>>(x, cluster, cmax, n);
  write_cx_kernel<<<gNF, B, 0, stream>>>(cmax, csize, o_cx, n);
  write_cb_cluster_kernel<<<gN, B, 0, stream>>>(cbmax, cluster, o_cb, o_cl, n);

  build_keys_kernel<<<gM, B, 0, stream>>>(row, col, mask, cluster, keys, E, n,
                                          M);
  for (int k = 2; k <= M; k <<= 1)
    for (int j = k >> 1; j > 0; j >>= 1)
      bitonic_kernel<<<gM, B, 0, stream>>>(keys, k, j);
  emit_edges_kernel<<<gE, B, 0, stream>>>(keys, o_er, o_ec, E, n);
}